// ClassificationLoss_85693187489836
// MI455X (gfx1250) — compile-verified
//
#include <hip/hip_runtime.h>
#include <hip/hip_bf16.h>
#include <math.h>

// ---------------- constants ----------------
#define NC      80
#define TOPK    10
#define GNUM    20
#define BATCH   32
#define EPSV    1e-9f
#define A_TOT   8400          // 6400 + 1600 + 400
#define GPAD    32            // padded GT column count for aligned stores
#define TILES_PER_B 525       // 400 + 100 + 25 tiles of 16 anchors
#define NEGINF  (-3.4e38f)

typedef float v2f __attribute__((ext_vector_type(2)));
typedef float v8f __attribute__((ext_vector_type(8)));

// ---------------------------------------------------------------------------
// Kernel 1: per 16-anchor tile (one wave32 per block):
//   - async-copy 16x80 cls tile global->LDS (GLOBAL_LOAD_ASYNC_TO_LDS_B128,
//     ASYNCcnt) while reg/anchor/GT loads issue in parallel
//   - row max + logsumexp (saved to ws), softmax scores in-place in LDS
//   - cls_score = scores @ onehot(labels) via V_WMMA_F32_16X16X4_F32 (x40)
//   - IoU vs 20 GT boxes, in_gts mask, metric = score * iou^6 -> ws plane
// ---------------------------------------------------------------------------
__global__ __launch_bounds__(32)
void tal_level_kernel(const float* __restrict__ cls0, const float* __restrict__ reg0,
                      const float* __restrict__ cls1, const float* __restrict__ reg1,
                      const float* __restrict__ cls2, const float* __restrict__ reg2,
                      const float* __restrict__ anchors,
                      const float* __restrict__ gt_boxes,
                      const int*   __restrict__ gt_labels,
                      float* __restrict__ metric,   // [B][A_TOT][GPAD]
                      float* __restrict__ lse)      // [B][A_TOT]
{
  __shared__ float sCls[16 * NC];   // scores tile (raw cls, then softmax'ed)
  __shared__ float sPd [16 * 4];    // predicted boxes
  __shared__ float sAnc[16 * 2];    // anchor centers (pixels)
  __shared__ float sGtB[GNUM * 4];
  __shared__ int   sGtL[32];

  const int lane = threadIdx.x;
  const int bid  = blockIdx.x;
  const int b    = bid / TILES_PER_B;
  const int t    = bid % TILES_PER_B;

  int lvl, a0;
  if (t < 400)      { lvl = 0; a0 = t * 16; }
  else if (t < 500) { lvl = 1; a0 = (t - 400) * 16; }
  else              { lvl = 2; a0 = (t - 500) * 16; }

  const float* clsp; const float* regp; int A; int offs; float stride;
  if (lvl == 0)      { clsp = cls0; regp = reg0; A = 6400; offs = 0;    stride = 8.f;  }
  else if (lvl == 1) { clsp = cls1; regp = reg1; A = 1600; offs = 6400; stride = 16.f; }
  else               { clsp = cls2; regp = reg2; A = 400;  offs = 8000; stride = 32.f; }

  // ---- async-copy the 16x80 cls tile directly into LDS (no VGPR staging) ---
  // Per the flat-aperture rule, the low 32 bits of a generic LDS pointer are
  // the workgroup-relative LDS byte offset expected in the async VDST VGPR.
  const float* src = clsp + ((size_t)b * A + a0) * NC;
  if (a0 + 16 < A) __builtin_prefetch(src + 16 * NC, 0, 1);  // global_prefetch_b8
  {
    const unsigned lds_base = (unsigned)(uintptr_t)(&sCls[0]);
    const unsigned long long gbase = (unsigned long long)(uintptr_t)src;
#pragma unroll
    for (int i = 0; i < 10; ++i) {
      const unsigned byteoff = (unsigned)(lane + 32 * i) * 16u;  // 320 x 16B
      const unsigned ldst = lds_base + byteoff;
      const unsigned long long ga = gbase + byteoff;
      asm volatile("global_load_async_to_lds_b128 %0, %1, off"
                   :: "v"(ldst), "v"(ga) : "memory");
    }
  }

  // GT labels / boxes for this batch (VMEM, overlaps the async copy)
  if (lane < GNUM) {
    sGtL[lane] = gt_labels[b * GNUM + lane];
    const float4 g4 = ((const float4*)gt_boxes)[(size_t)b * GNUM + lane];
    sGtB[lane * 4 + 0] = g4.x; sGtB[lane * 4 + 1] = g4.y;
    sGtB[lane * 4 + 2] = g4.z; sGtB[lane * 4 + 3] = g4.w;
  } else {
    sGtL[lane] = -1;
  }

  // predicted boxes + anchor centers for the 16 anchors
  if (lane < 16) {
    const int a = a0 + lane;
    const float4 r4 = ((const float4*)regp)[(size_t)b * A + a];
    const float ax = anchors[2 * (offs + a) + 0] * stride;
    const float ay = anchors[2 * (offs + a) + 1] * stride;
    sPd[lane * 4 + 0] = ax - r4.x * stride;
    sPd[lane * 4 + 1] = ay - r4.y * stride;
    sPd[lane * 4 + 2] = ax + r4.z * stride;
    sPd[lane * 4 + 3] = ay + r4.w * stride;
    sAnc[lane * 2 + 0] = ax;
    sAnc[lane * 2 + 1] = ay;
  }

  asm volatile("s_wait_asynccnt 0x0" ::: "memory");  // cls tile resident in LDS
  __syncthreads();

  // row softmax stats: lane (m, hi) covers classes [hi*40, hi*40+40)
  const int m  = lane & 15;
  const int hi = lane >> 4;
  float mx = NEGINF;
#pragma unroll 8
  for (int k = 0; k < 40; ++k) mx = fmaxf(mx, sCls[m * NC + hi * 40 + k]);
  float sm = 0.f;
#pragma unroll 8
  for (int k = 0; k < 40; ++k) sm += __expf(sCls[m * NC + hi * 40 + k] - mx);
  const float mxo = __shfl_xor(mx, 16, 32);
  const float smo = __shfl_xor(sm, 16, 32);
  const float M   = fmaxf(mx, mxo);
  const float S   = sm * __expf(mx - M) + smo * __expf(mxo - M);
  const float inv = 1.f / S;
  if (lane < 16) lse[(size_t)b * A_TOT + offs + a0 + lane] = M + __logf(S);

  // softmax in place (each lane rewrites exactly the range it owns)
#pragma unroll 8
  for (int k = 0; k < 40; ++k) {
    const int idx = m * NC + hi * 40 + k;
    sCls[idx] = __expf(sCls[idx] - M) * inv;
  }
  __syncthreads();

  float* mbase = metric + ((size_t)b * A_TOT + offs + a0) * GPAD;

  // two N-tiles cover 32 padded GT columns
#pragma unroll
  for (int nt = 0; nt < 2; ++nt) {
    const int col = (lane & 15) + nt * 16;
    const int lab = sGtL[col];                    // -1 for padded columns
    float gx1 = 0.f, gy1 = 0.f, gx2 = 0.f, gy2 = 0.f, garea = 0.f;
    if (col < GNUM) {
      gx1 = sGtB[col * 4 + 0]; gy1 = sGtB[col * 4 + 1];
      gx2 = sGtB[col * 4 + 2]; gy2 = sGtB[col * 4 + 3];
      garea = (gx2 - gx1) * (gy2 - gy1);
    }

    v8f c = {};
    // D = scores(16x80) @ onehot(80x16), 20 chunks of K=4
#pragma unroll
    for (int kc = 0; kc < 20; ++kc) {
      const int k0 = kc * 4 + hi * 2;             // this lane's K pair
      v2f av; av.x = sCls[m * NC + k0]; av.y = sCls[m * NC + k0 + 1];
      v2f bv; bv.x = (lab == k0) ? 1.f : 0.f; bv.y = (lab == k0 + 1) ? 1.f : 0.f;
      c = __builtin_amdgcn_wmma_f32_16x16x4_f32(false, av, false, bv,
                                                (short)0, c, false, false);
    }

    // per C-fragment element: row mr = r + 8*hi, column = col
#pragma unroll
    for (int r = 0; r < 8; ++r) {
      const int mr = r + 8 * hi;
      float val = 0.f;
      if (col < GNUM) {
        const float px1 = sPd[mr * 4 + 0], py1 = sPd[mr * 4 + 1];
        const float px2 = sPd[mr * 4 + 2], py2 = sPd[mr * 4 + 3];
        const float ix1 = fmaxf(px1, gx1), iy1 = fmaxf(py1, gy1);
        const float ix2 = fminf(px2, gx2), iy2 = fminf(py2, gy2);
        const float iw = fmaxf(ix2 - ix1, 0.f), ih = fmaxf(iy2 - iy1, 0.f);
        const float inter = iw * ih;
        const float pa = (px2 - px1) * (py2 - py1);
        const float iou = inter / (pa + garea - inter + EPSV);
        const float ax = sAnc[mr * 2 + 0], ay = sAnc[mr * 2 + 1];
        const bool ing = (ax - gx1 > EPSV) && (ay - gy1 > EPSV) &&
                         (gx2 - ax > EPSV) && (gy2 - ay > EPSV);
        const float i2 = iou * iou;
        const float i6 = i2 * i2 * i2;
        val = ing ? c[r] * i6 : 0.f;
      }
      mbase[(size_t)mr * GPAD + col] = val;       // pads written as 0
    }
  }
}

// ---------------------------------------------------------------------------
// Kernel 2: 10th-largest metric per (b, level, g). One wave per block.
// Per-lane sorted top-10 insertion, then ballot-merge across the wave.
// ---------------------------------------------------------------------------
__global__ __launch_bounds__(32)
void tal_topk_kernel(const float* __restrict__ metric, float* __restrict__ thresh)
{
  const int lane = threadIdx.x;
  const int bid  = blockIdx.x;            // b*60 + lvl*20 + g
  const int b = bid / 60, rem = bid % 60;
  const int lvl = rem / 20, g = rem % 20;

  int A, offs;
  if (lvl == 0)      { A = 6400; offs = 0;    }
  else if (lvl == 1) { A = 1600; offs = 6400; }
  else               { A = 400;  offs = 8000; }

  const float* base = metric + ((size_t)b * A_TOT + offs) * GPAD + g;

  float t[TOPK];
#pragma unroll
  for (int j = 0; j < TOPK; ++j) t[j] = NEGINF;

  for (int i = lane; i < A; i += 32) {
    const float v = base[(size_t)i * GPAD];
    if (v > t[TOPK - 1]) {
      t[TOPK - 1] = v;                    // single bubble pass (list was sorted)
#pragma unroll
      for (int j = TOPK - 1; j > 0; --j) {
        const float hi = fmaxf(t[j - 1], t[j]);
        const float lo = fminf(t[j - 1], t[j]);
        t[j - 1] = hi; t[j] = lo;
      }
    }
  }

  // merge 32 sorted lists: pop the global max TOPK times
  int   ptr = 0;
  float val = t[0];
  float kth = NEGINF;
#pragma unroll
  for (int round = 0; round < TOPK; ++round) {
    float mred = val;
#pragma unroll
    for (int off = 16; off > 0; off >>= 1)
      mred = fmaxf(mred, __shfl_xor(mred, off, 32));
    kth = mred;
    const unsigned long long msk = __ballot(val == mred);
    const int leader = __ffsll((long long)msk) - 1;
    if (lane == leader) {
      ++ptr;
      val = (ptr < TOPK) ? t[ptr] : NEGINF;
    }
  }
  thresh[bid] = kth;   // 10th largest
}

// ---------------------------------------------------------------------------
// Kernel 3: per-anchor assignment + CE accumulation.
// ---------------------------------------------------------------------------
__global__ void tal_assign_kernel(const float* __restrict__ cls0, const float* __restrict__ reg0,
                                  const float* __restrict__ cls1, const float* __restrict__ reg1,
                                  const float* __restrict__ cls2, const float* __restrict__ reg2,
                                  const float* __restrict__ anchors,
                                  const float* __restrict__ gt_boxes,
                                  const int*   __restrict__ gt_labels,
                                  const float* __restrict__ metric,
                                  const float* __restrict__ lse,
                                  const float* __restrict__ thresh,
                                  float* __restrict__ acc)
{
  const int tid = blockIdx.x * blockDim.x + threadIdx.x;
  if (tid >= BATCH * A_TOT) return;
  const int b = tid / A_TOT;
  const int a = tid % A_TOT;

  int lvl, A, offs; float stride;
  const float* clsp; const float* regp;
  if (a < 6400)      { lvl = 0; A = 6400; offs = 0;    stride = 8.f;  clsp = cls0; regp = reg0; }
  else if (a < 8000) { lvl = 1; A = 1600; offs = 6400; stride = 16.f; clsp = cls1; regp = reg1; }
  else               { lvl = 2; A = 400;  offs = 8000; stride = 32.f; clsp = cls2; regp = reg2; }
  const int al = a - offs;

  const float* th   = thresh + (b * 3 + lvl) * GNUM;
  const float* mrow = metric + (size_t)tid * GPAD;

  unsigned posmask = 0u;
  int cnt = 0;
#pragma unroll
  for (int g = 0; g < GNUM; ++g) {
    const float v = mrow[g];
    if (v > EPSV && v >= th[g]) { posmask |= (1u << g); ++cnt; }
  }
  if (cnt == 0) return;

  int gsel;
  if (cnt == 1) {
    gsel = __ffs(posmask) - 1;
  } else {
    // resolve multi-assignment by best IoU (ties -> lowest index)
    const float4 r4 = ((const float4*)regp)[(size_t)b * A + al];
    const float ax = anchors[2 * a + 0] * stride;
    const float ay = anchors[2 * a + 1] * stride;
    const float px1 = ax - r4.x * stride, py1 = ay - r4.y * stride;
    const float px2 = ax + r4.z * stride, py2 = ay + r4.w * stride;
    const float pa = (px2 - px1) * (py2 - py1);
    float best = -1.f; gsel = 0;
#pragma unroll
    for (int g = 0; g < GNUM; ++g) {
      if ((posmask >> g) & 1u) {
        const float4 g4 = ((const float4*)gt_boxes)[(size_t)b * GNUM + g];
        const float ix1 = fmaxf(px1, g4.x), iy1 = fmaxf(py1, g4.y);
        const float ix2 = fminf(px2, g4.z), iy2 = fminf(py2, g4.w);
        const float iw = fmaxf(ix2 - ix1, 0.f), ih = fmaxf(iy2 - iy1, 0.f);
        const float inter = iw * ih;
        const float ga = (g4.z - g4.x) * (g4.w - g4.y);
        const float iou = inter / (pa + ga - inter + EPSV);
        if (iou > best) { best = iou; gsel = g; }
      }
    }
  }

  const int lab = gt_labels[b * GNUM + gsel];
  const float clsval = clsp[((size_t)b * A + al) * NC + lab];
  const float ce = lse[tid] - clsval;    // -log_softmax[label]
  atomicAdd(&acc[0], ce);
  atomicAdd(&acc[1], 1.0f);
}

// ---------------------------------------------------------------------------
// Kernel 4: finalize loss = sum / max(n_pos, 1); emit (loss, n_pos).
// ---------------------------------------------------------------------------
__global__ void tal_finalize_kernel(const float* __restrict__ acc, float* __restrict__ out)
{
  const float s = acc[0];
  const float n = acc[1];
  out[0] = s / fmaxf(n, 1.0f);
  out[1] = n;
}

// ---------------------------------------------------------------------------
extern "C" void kernel_launch(void* const* d_in, const int* in_sizes, int n_in,
                              void* d_out, int out_size, void* d_ws, size_t ws_size,
                              hipStream_t stream) {
  // setup_inputs() dict order: cls0, reg0, cls1, reg1, cls2, reg2, anchors, gt_boxes, gt_labels
  const float* cls0 = (const float*)d_in[0];
  const float* reg0 = (const float*)d_in[1];
  const float* cls1 = (const float*)d_in[2];
  const float* reg1 = (const float*)d_in[3];
  const float* cls2 = (const float*)d_in[4];
  const float* reg2 = (const float*)d_in[5];
  const float* anch = (const float*)d_in[6];
  const float* gtb  = (const float*)d_in[7];
  const int*   gtl  = (const int*)d_in[8];
  float* out = (float*)d_out;

  float* W = (float*)d_ws;
  const size_t METRIC_ELEMS = (size_t)BATCH * A_TOT * GPAD;   // 8,601,600
  const size_t LSE_ELEMS    = (size_t)BATCH * A_TOT;          // 268,800
  const size_t TH_ELEMS     = (size_t)BATCH * 3 * GNUM;       // 1,920
  float* metric = W;
  float* lse    = metric + METRIC_ELEMS;
  float* thresh = lse + LSE_ELEMS;
  float* acc    = thresh + TH_ELEMS;

  hipMemsetAsync(acc, 0, 2 * sizeof(float), stream);

  tal_level_kernel<<<dim3(BATCH * TILES_PER_B), dim3(32), 0, stream>>>(
      cls0, reg0, cls1, reg1, cls2, reg2, anch, gtb, gtl, metric, lse);

  tal_topk_kernel<<<dim3(BATCH * 3 * GNUM), dim3(32), 0, stream>>>(metric, thresh);

  tal_assign_kernel<<<dim3((BATCH * A_TOT + 255) / 256), dim3(256), 0, stream>>>(
      cls0, reg0, cls1, reg1, cls2, reg2, anch, gtb, gtl, metric, lse, thresh, acc);

  tal_finalize_kernel<<<1, 1, 0, stream>>>(acc, out);
}